// MLPPredictor_E2E_3444563771452
// MI455X (gfx1250) — compile-verified
//
#include <hip/hip_runtime.h>
#include <hip/hip_bf16.h>
#include <math.h>

// ---------------------------------------------------------------------------
// MLPPredictor edge scorer, factored for MI455X (gfx1250, wave32).
//
// z[e] = LN(A[src] + B[dst] + polar@Wp + b1); out = relu(z)@W2 + b2
// where A[n] = [h,softmax(cls)][n] @ W1[0:128,:]   (precomputed per node)
//       B[n] = [h,softmax(cls)][n] @ W1[130:258,:] (precomputed per node)
//       Wp   = W1[128:130,:]
//
// FLOPs drop 5x vs the naive per-edge GEMM (2.6e10 vs 1.3e11); the node GEMM
// runs on V_WMMA_F32_16X16X4_F32 (fp32 tensor path — reference is fp32 and
// LayerNorm is variance-sensitive, so no downcast). The edge pass gathers
// 2KB/edge from a 205MB table (mostly resident in the 192MB L2); HBM traffic
// ~0.3GB ~ 13us @ 23.3TB/s. LDS tiles are filled with CDNA5 async
// global->LDS copies (ASYNCcnt) when the toolchain exposes the builtins.
// ---------------------------------------------------------------------------

#define HD     124
#define KDIM   128
#define HIDDEN 256
#define OUTC   512
#define M_TILE 32
#define N_TILE 64
#define XA_PAD 132   // 16B-aligned rows; 132%64=4 -> A-frag rows hit distinct banks
#define W_PAD  72    // 16B-aligned rows; 2*72%64=16 -> B-frag row pair disjoint banks
#define EPW    16    // edges per wave in the edge kernel

#if defined(__gfx1250__) && __has_builtin(__builtin_amdgcn_global_load_async_to_lds_b128)
#define USE_ASYNC_LDS 1
#else
#define USE_ASYNC_LDS 0
#endif

typedef __attribute__((ext_vector_type(2))) float v2f;
typedef __attribute__((ext_vector_type(4))) float f4;
typedef __attribute__((ext_vector_type(8))) float v8f;

__device__ __forceinline__ float dot4(f4 a, f4 b) {
    return a.x * b.x + a.y * b.y + a.z * b.z + a.w * b.w;
}
__device__ __forceinline__ f4 relu4(f4 v) {
    v.x = fmaxf(v.x, 0.f); v.y = fmaxf(v.y, 0.f);
    v.z = fmaxf(v.z, 0.f); v.w = fmaxf(v.w, 0.f);
    return v;
}
__device__ __forceinline__ float softmax1of4(const float* cl, int which) {
    float c0 = cl[0], c1 = cl[1], c2 = cl[2], c3 = cl[3];
    float mx = fmaxf(fmaxf(c0, c1), fmaxf(c2, c3));
    float e0 = __expf(c0 - mx), e1 = __expf(c1 - mx);
    float e2 = __expf(c2 - mx), e3 = __expf(c3 - mx);
    float ev = (which == 0) ? e0 : (which == 1) ? e1 : (which == 2) ? e2 : e3;
    return ev / (e0 + e1 + e2 + e3);
}

#if USE_ASYNC_LDS
// Builtin prototype (from hipcc diagnostic): args are pointers to a 16-byte
// int vector in the global (as1) / LDS (as3) address spaces.
typedef int b128_t __attribute__((vector_size(16)));
typedef __attribute__((address_space(1))) b128_t* gptr_b128;
typedef __attribute__((address_space(3))) b128_t* lptr_b128;

__device__ __forceinline__ void async_b128(const float* g, float* l) {
    __builtin_amdgcn_global_load_async_to_lds_b128(
        (gptr_b128)(g), (lptr_b128)(l), 0, 0);
}
__device__ __forceinline__ void wait_async0() {
#if __has_builtin(__builtin_amdgcn_s_wait_asynccnt)
    __builtin_amdgcn_s_wait_asynccnt(0);
#else
    asm volatile("s_wait_asynccnt 0x0" ::: "memory");
#endif
}
#endif

// --------------------------------------------------------------------------
// Kernel 1: per-node GEMM  AB[n, 0:512] = [h, softmax(cls)][n] @ Wab[128,512]
//   cols [0,256)   -> W1 rows 0..127     (src-block of W1)
//   cols [256,512) -> W1 rows 130..257   (dst-block of W1)
// Tile: 32 nodes x 64 cols per workgroup; 8 waves each own one 16x16 C tile.
// --------------------------------------------------------------------------
__global__ void __launch_bounds__(256)
node_gemm_wmma(const float* __restrict__ h, const float* __restrict__ cls,
               const float* __restrict__ W1, float* __restrict__ AB, int nNodes)
{
    __shared__ float xa[M_TILE * XA_PAD];   // node features, row-major [32][132]
    __shared__ float wt[KDIM * W_PAD];      // W tile, K-major [128][72]

    const int tid  = threadIdx.x;
    const int tile = blockIdx.x;
    const int m0 = (tile >> 3) * M_TILE;        // 512/64 = 8 col tiles
    const int n0 = (tile & 7) * N_TILE;

#if USE_ASYNC_LDS
    // h-part of xa: 32 rows x 31 aligned b128 chunks, async global->LDS.
    for (int i = tid; i < M_TILE * 31; i += 256) {
        const int r = i / 31, c4 = (i - r * 31) * 4;
        const int node = m0 + r;
        float* dst = &xa[r * XA_PAD + c4];
        if (node < nNodes) {
            async_b128(h + (size_t)node * HD + c4, dst);
        } else {
            *(f4*)dst = (f4)0.f;
        }
    }
    // softmax(cls) -> cols 124..127 (computed, regular ds_store).
    if (tid < M_TILE * 4) {
        const int r = tid >> 2, which = tid & 3;
        const int node = m0 + r;
        xa[r * XA_PAD + HD + which] =
            (node < nNodes) ? softmax1of4(cls + (size_t)node * 4, which) : 0.f;
    }
    // W tile: 128 rows x 16 aligned b128 chunks, async global->LDS.
    for (int i = tid; i < KDIM * 16; i += 256) {
        const int k = i >> 4, cc = (i & 15) * 4;
        const int j = n0 + cc;
        const int row = (j < HIDDEN) ? k : (130 + k);
        const int col = (j < HIDDEN) ? j : (j - HIDDEN);
        async_b128(W1 + (size_t)row * HIDDEN + col, &wt[k * W_PAD + cc]);
    }
    wait_async0();
#else
    for (int i = tid; i < M_TILE * KDIM; i += 256) {
        int r = i >> 7, c = i & 127;
        int node = m0 + r;
        float v = 0.f;
        if (node < nNodes) {
            if (c < HD) v = h[(size_t)node * HD + c];
            else        v = softmax1of4(cls + (size_t)node * 4, c - HD);
        }
        xa[r * XA_PAD + c] = v;
    }
    for (int i = tid; i < KDIM * N_TILE; i += 256) {
        int k = i >> 6, cc = i & 63;
        int j = n0 + cc;
        int row = (j < HIDDEN) ? k : (130 + k);
        int col = (j < HIDDEN) ? j : (j - HIDDEN);
        wt[k * W_PAD + cc] = W1[(size_t)row * HIDDEN + col];
    }
#endif
    __syncthreads();

    const int wave = tid >> 5, lane = tid & 31;
    const int wm = wave & 1;          // 2 waves along M (2*16 = 32 nodes)
    const int wn = wave >> 1;         // 4 waves along N (4*16 = 64 cols)
    const int ar = wm * 16 + (lane & 15);
    const int bc = wn * 16 + (lane & 15);
    const int khalf = 2 * (lane >> 4);    // lanes 16..31 hold K+2/K+3

    v8f acc = {};
    #pragma unroll 8
    for (int k = 0; k < KDIM; k += 4) {
        const int kk = k + khalf;
        v2f a, b;
        a.x = xa[ar * XA_PAD + kk];
        a.y = xa[ar * XA_PAD + kk + 1];
        b.x = wt[kk * W_PAD + bc];
        b.y = wt[(kk + 1) * W_PAD + bc];
        acc = __builtin_amdgcn_wmma_f32_16x16x4_f32(
                  false, a, false, b, (short)0, acc, false, false);
    }

    // C layout: VGPR v -> M = v (lanes 0-15) / v+8 (lanes 16-31); N = lane&15.
    const int mbase = m0 + wm * 16 + ((lane >> 4) ? 8 : 0);
    float* p = AB + (size_t)mbase * OUTC + (n0 + bc);
    if (mbase + 8 <= nNodes) {               // common case: straight b32 stores
        #pragma unroll
        for (int v = 0; v < 8; ++v) p[(size_t)v * OUTC] = acc[v];
    } else {
        #pragma unroll
        for (int v = 0; v < 8; ++v)
            if (mbase + v < nNodes) p[(size_t)v * OUTC] = acc[v];
    }
}

// --------------------------------------------------------------------------
// Shared tail: LayerNorm + ReLU + z@W2 + b2 for one edge (one wave32).
// Lane owns cols [4L,4L+4) and [128+4L,128+4L+4).
// --------------------------------------------------------------------------
__device__ __forceinline__ void ln_relu_head(
    f4 z0, f4 z1, f4 ga, f4 gb, f4 ba, f4 bb,
    f4 w2a0, f4 w2a1, f4 w2b0, f4 w2b1,
    float bias0, float bias1, float* __restrict__ out, long e, int lane)
{
    float s1 = z0.x + z0.y + z0.z + z0.w + z1.x + z1.y + z1.z + z1.w;
    float s2 = dot4(z0, z0) + dot4(z1, z1);
    #pragma unroll
    for (int m = 16; m >= 1; m >>= 1) {
        s1 += __shfl_xor(s1, m, 32);
        s2 += __shfl_xor(s2, m, 32);
    }
    const float mu  = s1 * (1.f / 256.f);
    const float var = s2 * (1.f / 256.f) - mu * mu;
    const float rs  = rsqrtf(var + 1e-5f);
    z0 = relu4((z0 - mu) * rs * ga + ba);
    z1 = relu4((z1 - mu) * rs * gb + bb);
    float o0 = dot4(z0, w2a0) + dot4(z1, w2b0);
    float o1 = dot4(z0, w2a1) + dot4(z1, w2b1);
    #pragma unroll
    for (int m = 16; m >= 1; m >>= 1) {
        o0 += __shfl_xor(o0, m, 32);
        o1 += __shfl_xor(o1, m, 32);
    }
    if (lane == 0) {
        out[e * 2 + 0] = o0 + bias0;
        out[e * 2 + 1] = o1 + bias1;
    }
}

// --------------------------------------------------------------------------
// Kernel 2: edge pass. One wave per edge, EPW edges/wave to amortize
// constant loads. Gathers are coalesced float4 reads of 1KB rows.
// --------------------------------------------------------------------------
__global__ void __launch_bounds__(256)
edge_kernel(const float* __restrict__ AB, const float* __restrict__ polar,
            const int* __restrict__ src, const int* __restrict__ dst,
            const float* __restrict__ W1, const float* __restrict__ b1,
            const float* __restrict__ gamma, const float* __restrict__ beta,
            const float* __restrict__ W2, const float* __restrict__ b2,
            float* __restrict__ out, int E)
{
    const int lane = threadIdx.x & 31;
    const long e0 = ((long)blockIdx.x * 8 + (threadIdx.x >> 5)) * EPW;
    const int c0 = 4 * lane, c1 = 128 + 4 * lane;

    // Per-wave constants (uniform; hit L0/L2).
    const f4* Wp0 = (const f4*)(W1 + (size_t)128 * HIDDEN);
    const f4* Wp1 = (const f4*)(W1 + (size_t)129 * HIDDEN);
    const f4 wp0a = Wp0[lane], wp0b = Wp0[lane + 32];
    const f4 wp1a = Wp1[lane], wp1b = Wp1[lane + 32];
    const f4 b1a = ((const f4*)b1)[lane],    b1b = ((const f4*)b1)[lane + 32];
    const f4 ga  = ((const f4*)gamma)[lane], gb  = ((const f4*)gamma)[lane + 32];
    const f4 ba  = ((const f4*)beta)[lane],  bb  = ((const f4*)beta)[lane + 32];
    f4 w2a0, w2a1, w2b0, w2b1;
    #pragma unroll
    for (int i = 0; i < 4; ++i) {
        w2a0[i] = W2[(c0 + i) * 2 + 0]; w2a1[i] = W2[(c0 + i) * 2 + 1];
        w2b0[i] = W2[(c1 + i) * 2 + 0]; w2b1[i] = W2[(c1 + i) * 2 + 1];
    }
    const float bias0 = b2[0], bias1 = b2[1];

    for (int t = 0; t < EPW; ++t) {
        const long e = e0 + t;
        if (e >= E) break;
        const int s = src[e], d = dst[e];
        const f4* Ar = (const f4*)(AB + (size_t)s * OUTC);           // A part
        const f4* Br = (const f4*)(AB + (size_t)d * OUTC + HIDDEN);  // B part
        const float p0 = polar[e * 2], p1 = polar[e * 2 + 1];
        f4 z0 = Ar[lane]      + Br[lane]      + p0 * wp0a + p1 * wp1a + b1a;
        f4 z1 = Ar[lane + 32] + Br[lane + 32] + p0 * wp0b + p1 * wp1b + b1b;
        ln_relu_head(z0, z1, ga, gb, ba, bb, w2a0, w2a1, w2b0, w2b1,
                     bias0, bias1, out, e, lane);
    }
}

// --------------------------------------------------------------------------
// Fallback (only if workspace < N*512*4 bytes): fully fused per-edge kernel.
// One wave per edge; stages the 258-feature vector in LDS and streams W1.
// --------------------------------------------------------------------------
__global__ void __launch_bounds__(256)
edge_fused_fallback(const float* __restrict__ h, const float* __restrict__ cls,
                    const float* __restrict__ polar,
                    const int* __restrict__ src, const int* __restrict__ dst,
                    const float* __restrict__ W1, const float* __restrict__ b1,
                    const float* __restrict__ gamma, const float* __restrict__ beta,
                    const float* __restrict__ W2, const float* __restrict__ b2,
                    float* __restrict__ out, int E)
{
    __shared__ float xs[8][260];
    const int wave = threadIdx.x >> 5, lane = threadIdx.x & 31;
    const long e = (long)blockIdx.x * 8 + wave;
    const bool active = (e < E);
    float* x = xs[wave];
    int s = 0, d = 0;
    if (active) {
        s = src[e]; d = dst[e];
        for (int i = lane; i < HD; i += 32) {
            x[i]       = h[(size_t)s * HD + i];
            x[130 + i] = h[(size_t)d * HD + i];
        }
        if (lane < 4) {
            x[124 + lane] = softmax1of4(cls + (size_t)s * 4, lane);
            x[254 + lane] = softmax1of4(cls + (size_t)d * 4, lane);
        }
        if (lane < 2) x[128 + lane] = polar[e * 2 + lane];
    }
    __syncthreads();
    if (!active) return;

    const int c0 = 4 * lane, c1 = 128 + 4 * lane;
    f4 z0 = ((const f4*)b1)[lane], z1 = ((const f4*)b1)[lane + 32];
    for (int k = 0; k < 258; ++k) {
        const float xv = x[k];
        const f4* Wk = (const f4*)(W1 + (size_t)k * HIDDEN);
        z0 += xv * Wk[lane];
        z1 += xv * Wk[lane + 32];
    }
    const f4 ga = ((const f4*)gamma)[lane], gb = ((const f4*)gamma)[lane + 32];
    const f4 ba = ((const f4*)beta)[lane],  bb = ((const f4*)beta)[lane + 32];
    f4 w2a0, w2a1, w2b0, w2b1;
    #pragma unroll
    for (int i = 0; i < 4; ++i) {
        w2a0[i] = W2[(c0 + i) * 2 + 0]; w2a1[i] = W2[(c0 + i) * 2 + 1];
        w2b0[i] = W2[(c1 + i) * 2 + 0]; w2b1[i] = W2[(c1 + i) * 2 + 1];
    }
    ln_relu_head(z0, z1, ga, gb, ba, bb, w2a0, w2a1, w2b0, w2b1,
                 b2[0], b2[1], out, e, lane);
}

extern "C" void kernel_launch(void* const* d_in, const int* in_sizes, int n_in,
                              void* d_out, int out_size, void* d_ws, size_t ws_size,
                              hipStream_t stream)
{
    const float* h     = (const float*)d_in[0];
    const float* cls   = (const float*)d_in[1];
    const float* polar = (const float*)d_in[2];
    const int*   src   = (const int*)d_in[3];
    const int*   dst   = (const int*)d_in[4];
    const float* W1    = (const float*)d_in[5];
    const float* b1    = (const float*)d_in[6];
    const float* gamma = (const float*)d_in[7];
    const float* beta  = (const float*)d_in[8];
    const float* W2    = (const float*)d_in[9];
    const float* b2    = (const float*)d_in[10];
    float* out = (float*)d_out;

    const int nNodes = in_sizes[0] / HD;   // 100000
    const int E      = in_sizes[3];        // 1000000
    const size_t ab_bytes = (size_t)nNodes * OUTC * sizeof(float); // ~205 MB

    if (ws_size >= ab_bytes) {
        float* AB = (float*)d_ws;
        const int mtiles = (nNodes + M_TILE - 1) / M_TILE;
        node_gemm_wmma<<<mtiles * 8, 256, 0, stream>>>(h, cls, W1, AB, nNodes);
        const int eblocks = (E + 8 * EPW - 1) / (8 * EPW);
        edge_kernel<<<eblocks, 256, 0, stream>>>(AB, polar, src, dst, W1, b1,
                                                 gamma, beta, W2, b2, out, E);
    } else {
        const int eblocks = (E + 7) / 8;
        edge_fused_fallback<<<eblocks, 256, 0, stream>>>(h, cls, polar, src, dst,
                                                         W1, b1, gamma, beta,
                                                         W2, b2, out, E);
    }
}